// GNN_pyg_45904610459949
// MI455X (gfx1250) — compile-verified
//
#include <hip/hip_runtime.h>
#include <hip/hip_bf16.h>
#include <math.h>

#define N_NODES  50000
#define N_EDGES  800000
#define D_IN     64
#define WIDTH    128
#define N_GRAPHS 256

typedef __attribute__((ext_vector_type(2))) float v2f;
typedef __attribute__((ext_vector_type(8))) float v8f;

// ---------------------------------------------------------------- utilities
__global__ void zero_kernel(float* __restrict__ p, long long n) {
    long long i = (long long)blockIdx.x * blockDim.x + threadIdx.x;
    if (i < n) p[i] = 0.0f;
}

// ------------------------------------------------- step 1: s[src] += x[dst]
// one wave (32 lanes) per edge, 2 floats per lane (D_IN = 64)
__global__ void scatter_feat_kernel(const float* __restrict__ x,
                                    const int* __restrict__ src,
                                    const int* __restrict__ dst,
                                    float* __restrict__ s) {
    int wave = (int)((blockIdx.x * blockDim.x + threadIdx.x) >> 5);
    int lane = threadIdx.x & 31;
    if (wave >= N_EDGES) return;
    int si = src[wave], di = dst[wave];
    const float2 v = *(const float2*)(x + (size_t)di * D_IN + lane * 2);
    float* sr = s + (size_t)si * D_IN + lane * 2;
    atomicAdd(sr + 0, v.x);
    atomicAdd(sr + 1, v.y);
}

// row-wise L2 normalize, one wave per node
__global__ void normalize_kernel(float* __restrict__ s) {
    int wave = (int)((blockIdx.x * blockDim.x + threadIdx.x) >> 5);
    int lane = threadIdx.x & 31;
    if (wave >= N_NODES) return;
    float* row = s + (size_t)wave * D_IN + lane * 2;
    float2 v = *(float2*)row;
    float ss = v.x * v.x + v.y * v.y;
    #pragma unroll
    for (int m = 16; m >= 1; m >>= 1) ss += __shfl_xor(ss, m, 32);
    float inv = 1.0f / sqrtf(ss);   // norm==0 -> inf*0 = NaN, same as reference
    v.x *= inv; v.y *= inv;
    *(float2*)row = v;
}

// ----------------------------------------------------- degrees / dis factor
__global__ void degree_kernel(const int* __restrict__ dst, float* __restrict__ deg) {
    int e = blockIdx.x * blockDim.x + threadIdx.x;
    if (e < N_EDGES) atomicAdd(&deg[dst[e]], 1.0f);
}

__global__ void dis_kernel(float* __restrict__ deg) {
    int i = blockIdx.x * blockDim.x + threadIdx.x;
    if (i < N_NODES) deg[i] = rsqrtf(deg[i] + 1.0f);   // self loop: deg >= 1
}

// ------------------------------------------------------- WMMA fp32 GEMM
// C[M x 128] = A[M x K] * B[K x 128], M = 50000 (3125 * 16), K in {64,128}
// block = 256 threads = 8 waves; wave w computes 16x16 tile at cols w*16.
// B chunk (64 x 128) staged in LDS with stride 132 (pad 4) for bank spread.
__global__ __launch_bounds__(256) void gemm_wmma_kernel(
        const float* __restrict__ A, const float* __restrict__ B,
        float* __restrict__ C, int K) {
    __shared__ float Bs[64 * 132];
    const int wave = threadIdx.x >> 5;
    const int lane = threadIdx.x & 31;
    const int half = lane >> 4;       // 0: k pair {0,1}, 1: k pair {2,3}
    const int l16  = lane & 15;
    const int m0   = blockIdx.x * 16;
    const int n0   = wave * 16;

    const float* arow = A + (size_t)(m0 + l16) * K + 2 * half;
    v8f acc = {};

    for (int k0 = 0; k0 < K; k0 += 64) {
        // cooperative stage of 64 x 128 slice of B into LDS
        for (int t = threadIdx.x; t < 64 * 128; t += 256) {
            int kk = t >> 7, cc = t & 127;
            Bs[kk * 132 + cc] = B[(size_t)(k0 + kk) * WIDTH + cc];
        }
        __syncthreads();

        #pragma unroll 4
        for (int kk = 0; kk < 64; kk += 4) {
            // A 16x4 fp32: lane(0-15)=rows, VGPR0/1 = k, k+1; lanes 16-31: k+2,k+3
            const float2 av = *(const float2*)(arow + k0 + kk);
            v2f a; a.x = av.x; a.y = av.y;
            // B 4x16 fp32 mirrored layout from LDS
            const int kb = kk + 2 * half;
            v2f b;
            b.x = Bs[kb * 132 + n0 + l16];
            b.y = Bs[(kb + 1) * 132 + n0 + l16];
            acc = __builtin_amdgcn_wmma_f32_16x16x4_f32(
                false, a, false, b, (short)0, acc, false, false);
        }
        __syncthreads();
    }

    // D 16x16 fp32: VGPR v -> row m0 + v + 8*half, col n0 + l16
    float* crow = C + (size_t)(m0 + 8 * half) * WIDTH + n0 + l16;
    #pragma unroll
    for (int v = 0; v < 8; ++v) crow[(size_t)v * WIDTH] = acc[v];
}

// --------------------------------------- edge messages: accum[dst] += n*hW[src]
// one wave per edge, float4 per lane (WIDTH = 128 = 32 lanes * 4)
__global__ void edge_msg_kernel(const int* __restrict__ src,
                                const int* __restrict__ dst,
                                const float* __restrict__ dis,
                                const float* __restrict__ hW,
                                float* __restrict__ accum) {
    int wave = (int)((blockIdx.x * blockDim.x + threadIdx.x) >> 5);
    int lane = threadIdx.x & 31;
    if (wave >= N_EDGES) return;
    int si = src[wave], di = dst[wave];
    float coef = dis[si] * dis[di];
    const float4 v = *(const float4*)(hW + (size_t)si * WIDTH + lane * 4);
    float* orow = accum + (size_t)di * WIDTH + lane * 4;
    atomicAdd(orow + 0, coef * v.x);
    atomicAdd(orow + 1, coef * v.y);
    atomicAdd(orow + 2, coef * v.z);
    atomicAdd(orow + 3, coef * v.w);
}

// self-loop term + bias + relu (fused, no atomics needed: deterministic per i)
__global__ void finalize_kernel(const float* __restrict__ accum,
                                const float* __restrict__ hW,
                                const float* __restrict__ dis,
                                const float* __restrict__ bias,
                                float* __restrict__ hout) {
    long long idx = (long long)blockIdx.x * blockDim.x + threadIdx.x;
    if (idx >= (long long)N_NODES * WIDTH) return;
    int i = (int)(idx >> 7), c = (int)(idx & 127);
    float d = dis[i];
    float val = accum[idx] + d * d * hW[idx] + bias[c];
    hout[idx] = fmaxf(val, 0.0f);
}

// ----------------------------------------------------------- mean pool + lin
__global__ void pool_kernel(const float* __restrict__ h,
                            const int* __restrict__ batch,
                            float* __restrict__ sums,
                            float* __restrict__ cnt) {
    int wave = (int)((blockIdx.x * blockDim.x + threadIdx.x) >> 5);
    int lane = threadIdx.x & 31;
    if (wave >= N_NODES) return;
    int g = batch[wave];
    const float4 v = *(const float4*)(h + (size_t)wave * WIDTH + lane * 4);
    float* srow = sums + (size_t)g * WIDTH + lane * 4;
    atomicAdd(srow + 0, v.x);
    atomicAdd(srow + 1, v.y);
    atomicAdd(srow + 2, v.z);
    atomicAdd(srow + 3, v.w);
    if (lane == 0) atomicAdd(&cnt[g], 1.0f);
}

__global__ void final_lin_kernel(const float* __restrict__ sums,
                                 const float* __restrict__ cnt,
                                 const float* __restrict__ lw,
                                 const float* __restrict__ lb,
                                 float* __restrict__ out) {
    int wave = (int)((blockIdx.x * blockDim.x + threadIdx.x) >> 5);
    int lane = threadIdx.x & 31;
    if (wave >= N_GRAPHS) return;
    const float* srow = sums + (size_t)wave * WIDTH + lane * 4;
    const float* wrow = lw + lane * 4;
    float d = srow[0]*wrow[0] + srow[1]*wrow[1] + srow[2]*wrow[2] + srow[3]*wrow[3];
    #pragma unroll
    for (int m = 16; m >= 1; m >>= 1) d += __shfl_xor(d, m, 32);
    if (lane == 0) out[wave] = d / fmaxf(cnt[wave], 1.0f) + lb[0];
}

// --------------------------------------------------------------------- host
extern "C" void kernel_launch(void* const* d_in, const int* in_sizes, int n_in,
                              void* d_out, int out_size, void* d_ws, size_t ws_size,
                              hipStream_t stream) {
    const float* x    = (const float*)d_in[0];
    const int*   eidx = (const int*)  d_in[1];
    const int*   src  = eidx;
    const int*   dst  = eidx + N_EDGES;
    const int*   batch= (const int*)  d_in[2];
    const float* W0   = (const float*)d_in[3];
    const float* b0   = (const float*)d_in[4];
    const float* W1   = (const float*)d_in[5];
    const float* b1   = (const float*)d_in[6];
    const float* W2   = (const float*)d_in[7];
    const float* b2   = (const float*)d_in[8];
    const float* lw   = (const float*)d_in[9];
    const float* lb   = (const float*)d_in[10];
    float* out = (float*)d_out;

    // workspace layout (floats); total ~22.5M floats (~90 MB)
    float* ws    = (float*)d_ws;
    size_t off   = 0;
    float* s0    = ws + off; off += (size_t)N_NODES * D_IN;   // [N,64] feat aggr / h0
    float* dis   = ws + off; off += (size_t)N_NODES;          // deg -> dis
    float* hA    = ws + off; off += (size_t)N_NODES * WIDTH;  // current hidden
    float* hW    = ws + off; off += (size_t)N_NODES * WIDTH;  // h @ W
    float* accum = ws + off; off += (size_t)N_NODES * WIDTH;  // edge aggregation
    float* psum  = ws + off; off += (size_t)N_GRAPHS * WIDTH; // pooled sums
    float* pcnt  = ws + off; off += (size_t)N_GRAPHS;         // pooled counts

    const int B = 256;
    const int gEdgeWave = (N_EDGES * 32) / B;            // 100000
    const int gNodeWave = (N_NODES * 32) / B;            // 6250
    const int gNW       = (N_NODES * WIDTH) / B;         // 25000
    const int gGemm     = N_NODES / 16;                  // 3125

    // step 1: s[src] += x[dst]; L2 normalize
    zero_kernel<<<(N_NODES * D_IN) / B, B, 0, stream>>>(s0, (long long)N_NODES * D_IN);
    scatter_feat_kernel<<<gEdgeWave, B, 0, stream>>>(x, src, dst, s0);
    normalize_kernel<<<gNodeWave, B, 0, stream>>>(s0);

    // degree -> dis = rsqrt(indeg + 1)
    zero_kernel<<<(N_NODES + B - 1) / B, B, 0, stream>>>(dis, N_NODES);
    degree_kernel<<<(N_EDGES + B - 1) / B, B, 0, stream>>>(dst, dis);
    dis_kernel<<<(N_NODES + B - 1) / B, B, 0, stream>>>(dis);

    // three GCN layers
    const float* hin = s0;
    int K = D_IN;
    const float* Ws[3] = {W0, W1, W2};
    const float* bs[3] = {b0, b1, b2};
    for (int l = 0; l < 3; ++l) {
        gemm_wmma_kernel<<<gGemm, B, 0, stream>>>(hin, Ws[l], hW, K);
        zero_kernel<<<gNW, B, 0, stream>>>(accum, (long long)N_NODES * WIDTH);
        edge_msg_kernel<<<gEdgeWave, B, 0, stream>>>(src, dst, dis, hW, accum);
        finalize_kernel<<<gNW, B, 0, stream>>>(accum, hW, dis, bs[l], hA);
        hin = hA;
        K = WIDTH;
    }

    // mean pool + final linear
    zero_kernel<<<((N_GRAPHS * WIDTH + N_GRAPHS) + B - 1) / B, B, 0, stream>>>(
        psum, (long long)N_GRAPHS * WIDTH + N_GRAPHS);   // psum and pcnt contiguous
    pool_kernel<<<gNodeWave, B, 0, stream>>>(hA, batch, psum, pcnt);
    final_lin_kernel<<<(N_GRAPHS * 32) / B, B, 0, stream>>>(psum, pcnt, lw, lb, out);
}